// PatchNCELoss_62466004353592
// MI455X (gfx1250) — compile-verified
//
#include <hip/hip_runtime.h>
#include <math.h>

// ---------------------------------------------------------------------------
// PatchNCE loss on gfx1250 (MI455X).
//   feat_q, feat_k : [4,128,32,32] f32 ; neg_indices : [4096,128] i32
//   out : scalar f32 loss
// Strategy: normalize+transpose to [N,C] row-major, then per-query WMMA
// (V_WMMA_F32_16X16X4_F32) with the key gather loaded directly into the
// B-matrix VGPR layout via one global_load_b64 per lane per K-chunk.
// ---------------------------------------------------------------------------

typedef __attribute__((ext_vector_type(2))) float v2f;
typedef __attribute__((ext_vector_type(8))) float v8f;

#define C_DIM   128
#define HW      1024          // 32*32
#define N_TOTAL 4096          // 4*HW
#define K_NEG   128
#define INV_T   (1.0f / 0.07f)
#define L2_EPS  1e-12f

// ---- Kernel 1: L2-normalize over channel dim and transpose to [N, C] -------
__global__ __launch_bounds__(256) void pnce_normalize(
    const float* __restrict__ q, const float* __restrict__ k,
    float* __restrict__ qn, float* __restrict__ kn) {
  int n = blockIdx.x * 256 + threadIdx.x;
  if (n >= N_TOTAL) return;
  int b  = n >> 10;          // n / HW
  int hw = n & (HW - 1);     // n % HW
  const float* srcs[2] = {q, k};
  float*       dsts[2] = {qn, kn};
  for (int s = 0; s < 2; ++s) {
    const float* base = srcs[s] + (size_t)b * C_DIM * HW + hw;  // stride HW over c
    float ss = 0.0f;
#pragma unroll 8
    for (int c = 0; c < C_DIM; ++c) {
      float v = base[c * HW];
      ss += v * v;
    }
    float rn = 1.0f / fmaxf(sqrtf(ss), L2_EPS);
    float* out = dsts[s] + (size_t)n * C_DIM;
#pragma unroll 8
    for (int c = 0; c < C_DIM; ++c) out[c] = base[c * HW] * rn;
  }
}

// ---- Kernel 2: per-query logits via WMMA + logsumexp -----------------------
// Block = 256 threads = 8 waves; block b handles query n = b.
// Wave w computes the 16 logits for negatives [16w, 16w+16) as one 16x16
// WMMA tile accumulated over 32 chunks of the C=128 contraction.
__global__ __launch_bounds__(256) void pnce_main(
    const float* __restrict__ qn, const float* __restrict__ kn,
    const int* __restrict__ negidx, float* __restrict__ lossbuf) {
  __shared__ float s_logits[K_NEG];

  const int n     = blockIdx.x;
  const int lane  = threadIdx.x & 31;
  const int wave  = threadIdx.x >> 5;
  const int col   = lane & 15;          // B-matrix column owned by this lane
  const int khalf = (lane >> 4) << 1;   // lanes 0-15 -> K=0,1 ; lanes 16-31 -> K=2,3

  // Gathered key row for this lane's column (f32 K-pairs land in B layout).
  const int key = negidx[n * K_NEG + wave * 16 + col];
  const float* bptr = kn + (size_t)key * C_DIM + khalf;
  const float* aptr = qn + (size_t)n   * C_DIM + khalf;   // broadcast query row

  v8f acc = {};
#pragma unroll
  for (int c0 = 0; c0 < C_DIM; c0 += 4) {
    v2f a2 = *(const v2f*)(aptr + c0);   // A 16x4: all rows = fq[n]
    v2f b2 = *(const v2f*)(bptr + c0);   // B 4x16: column j = gathered key j
    // 8 args: (neg_a, A, neg_b, B, c_mod, C, reuse_a, reuse_b)
    acc = __builtin_amdgcn_wmma_f32_16x16x4_f32(
        false, a2, false, b2, (short)0, acc, false, false);
  }
  // D row 0 (lanes 0-15, VGPR0) holds the 16 logits (all rows identical).
  if (lane < 16) s_logits[wave * 16 + lane] = acc[0];

  // Positive pair dot by wave 0 (each lane: 4 channels, then butterfly sum).
  float pos = 0.0f;
  if (wave == 0) {
    const float* qr = qn + (size_t)n * C_DIM + lane * 4;
    const float* kr = kn + (size_t)n * C_DIM + lane * 4;
    float p = 0.0f;
#pragma unroll
    for (int i = 0; i < 4; ++i) p += qr[i] * kr[i];
#pragma unroll
    for (int off = 16; off >= 1; off >>= 1) p += __shfl_xor(p, off, 32);
    pos = p * INV_T;                     // all lanes of wave 0 hold it
  }
  __syncthreads();

  // Wave 0: 129-way logsumexp (128 negatives + positive), write per-query loss.
  if (wave == 0) {
    float v[4];
    float m = pos;
#pragma unroll
    for (int i = 0; i < 4; ++i) {
      v[i] = s_logits[lane * 4 + i] * INV_T;
      m = fmaxf(m, v[i]);
    }
#pragma unroll
    for (int off = 16; off >= 1; off >>= 1) m = fmaxf(m, __shfl_xor(m, off, 32));
    float s = 0.0f;
#pragma unroll
    for (int i = 0; i < 4; ++i) s += expf(v[i] - m);
    if (lane == 0) s += expf(pos - m);   // positive term counted once
#pragma unroll
    for (int off = 16; off >= 1; off >>= 1) s += __shfl_xor(s, off, 32);
    if (lane == 0) lossbuf[n] = m + logf(s) - pos;
  }
}

// ---- Kernel 3: deterministic mean over 4096 per-query losses ---------------
__global__ __launch_bounds__(256) void pnce_reduce(
    const float* __restrict__ lossbuf, float* __restrict__ out) {
  __shared__ float s[256];
  float t = 0.0f;
  for (int i = threadIdx.x; i < N_TOTAL; i += 256) t += lossbuf[i];
  s[threadIdx.x] = t;
  __syncthreads();
  for (int off = 128; off > 0; off >>= 1) {
    if ((int)threadIdx.x < off) s[threadIdx.x] += s[threadIdx.x + off];
    __syncthreads();
  }
  if (threadIdx.x == 0) out[0] = s[0] / (float)N_TOTAL;
}

extern "C" void kernel_launch(void* const* d_in, const int* in_sizes, int n_in,
                              void* d_out, int out_size, void* d_ws, size_t ws_size,
                              hipStream_t stream) {
  (void)in_sizes; (void)n_in; (void)out_size; (void)ws_size;
  const float* fq  = (const float*)d_in[0];
  const float* fk  = (const float*)d_in[1];
  const int*   neg = (const int*)d_in[2];

  // Workspace: [qn 2MB][kn 2MB][per-query loss 16KB]  (~4.02 MB total)
  float* qn = (float*)d_ws;
  float* kn = qn + (size_t)N_TOTAL * C_DIM;
  float* lb = kn + (size_t)N_TOTAL * C_DIM;

  pnce_normalize<<<N_TOTAL / 256, 256, 0, stream>>>(fq, fk, qn, kn);
  pnce_main<<<N_TOTAL, 256, 0, stream>>>(qn, kn, neg, lb);
  pnce_reduce<<<1, 256, 0, stream>>>(lb, (float*)d_out);
}